// GCN_26903675142314
// MI455X (gfx1250) — compile-verified
//
#include <hip/hip_runtime.h>
#include <hip/hip_bf16.h>

typedef __attribute__((ext_vector_type(2))) float v2f;
typedef __attribute__((ext_vector_type(8))) float v8f;

#define NEG_SLOPE 0.2f
#define ACT_SLOPE 0.01f
#define FLT_MAX_F 3.402823466e38f

__device__ __forceinline__ float leaky(float x, float s) { return x >= 0.f ? x : s * x; }

__device__ __forceinline__ float atomicMaxF(float* addr, float val) {
    int* ai = (int*)addr;
    int old = __float_as_int(*addr);
    while (__int_as_float(old) < val) {
        int assumed = old;
        old = atomicCAS(ai, assumed, __float_as_int(val));
        if (old == assumed) break;
    }
    return __int_as_float(old);
}

// ---------------------------------------------------------------------------
// fp32 WMMA GEMM: C[M,N] = A[M,K] * W[K,N], all row-major.
// One wave per 16x16 output tile, K stepped by 4 via V_WMMA_F32_16X16X4_F32.
// Requires M%16==0, N%16==0, K%4==0.
// A frag (16x4): lanes 0-15 hold K={k0,k0+1}, lanes 16-31 hold K={k0+2,k0+3}, row = lane%16.
// B frag (4x16): same K split per lane-half, col = lane%16.
// C frag: VGPR v -> row = v + (lane/16)*8, col = lane%16.
// ---------------------------------------------------------------------------
__global__ void gemm_wmma_f32(const float* __restrict__ A, const float* __restrict__ W,
                              float* __restrict__ C, int M, int N, int K) {
    int wave = (int)((blockIdx.x * blockDim.x + threadIdx.x) >> 5);
    int ntn = N >> 4;
    int ntiles = (M >> 4) * ntn;
    if (wave >= ntiles) return;               // wave-uniform branch; EXEC stays all-1
    int tm = wave / ntn, tn = wave % ntn;
    int lane = threadIdx.x & 31;
    int r = lane & 15;
    int half = lane >> 4;

    const float* arow = A + (size_t)(tm * 16 + r) * K;
    const float* wcol = W + tn * 16 + r;

    v8f acc = {};
    for (int k0 = 0; k0 < K; k0 += 4) {
        v2f a, b;
        a.x = arow[k0 + 2 * half];
        a.y = arow[k0 + 2 * half + 1];
        b.x = wcol[(size_t)(k0 + 2 * half) * N];
        b.y = wcol[(size_t)(k0 + 2 * half + 1) * N];
        acc = __builtin_amdgcn_wmma_f32_16x16x4_f32(false, a, false, b, (short)0, acc,
                                                    false, false);
    }
    float* crow = C + (size_t)(tm * 16 + half * 8) * N + tn * 16 + r;
#pragma unroll
    for (int v = 0; v < 8; ++v) crow[(size_t)v * N] = acc[v];
}

// Per-node attention logits: e_src[n] = h[n,:]·as, e_dst[n] = h[n,:]·ad. Wave per node.
__global__ void dot2_kernel(const float* __restrict__ h, const float* __restrict__ as,
                            const float* __restrict__ ad, float* __restrict__ esrc,
                            float* __restrict__ edst, int N, int F) {
    int wave = (int)((blockIdx.x * blockDim.x + threadIdx.x) >> 5);
    if (wave >= N) return;
    int lane = threadIdx.x & 31;
    const float* row = h + (size_t)wave * F;
    float s0 = 0.f, s1 = 0.f;
    for (int f = lane; f < F; f += 32) {
        float v = row[f];
        s0 += v * as[f];
        s1 += v * ad[f];
    }
#pragma unroll
    for (int off = 16; off > 0; off >>= 1) {
        s0 += __shfl_down(s0, off, 32);
        s1 += __shfl_down(s1, off, 32);
    }
    if (lane == 0) { esrc[wave] = s0; edst[wave] = s1; }
}

__global__ void fill_kernel(float* __restrict__ p, float v, long n) {
    long i = blockIdx.x * (long)blockDim.x + threadIdx.x;
    if (i < n) p[i] = v;
}

// Pass 1 over edges (incl. self-loops at index >= E): e = leaky(es[src]+ed[dst]); max into m[dst].
__global__ void edge_emax(const int* __restrict__ ei, int E, int N,
                          const float* __restrict__ esrc, const float* __restrict__ edst,
                          float* __restrict__ e_out, float* __restrict__ m) {
    int i = blockIdx.x * blockDim.x + threadIdx.x;
    int T = E + N;
    if (i >= T) return;
    int s, d;
    if (i < E) { s = ei[i]; d = ei[E + i]; } else { s = d = i - E; }
    float e = leaky(esrc[s] + edst[d], NEG_SLOPE);
    e_out[i] = e;
    atomicMaxF(&m[d], e);
}

// Pass 2: w = exp(e - m[dst]); denom[dst] += w.
__global__ void edge_expsum(const int* __restrict__ ei, int E, int N,
                            const float* __restrict__ m, float* __restrict__ w,
                            float* __restrict__ denom) {
    int i = blockIdx.x * blockDim.x + threadIdx.x;
    int T = E + N;
    if (i >= T) return;
    int d = (i < E) ? ei[E + i] : i - E;
    float ww = expf(w[i] - m[d]);
    w[i] = ww;
    atomicAdd(&denom[d], ww);
}

// Pass 3 (wide features): wave per edge, lanes cover channels. out[dst,:] += alpha * h[src,:].
__global__ void aggregate_vec(const int* __restrict__ ei, int E, int N,
                              const float* __restrict__ w, const float* __restrict__ denom,
                              const float* __restrict__ h, float* __restrict__ out, int F) {
    int wave = (int)((blockIdx.x * blockDim.x + threadIdx.x) >> 5);
    int T = E + N;
    if (wave >= T) return;
    int lane = threadIdx.x & 31;
    int s, d;
    if (wave < E) { s = ei[wave]; d = ei[E + wave]; } else { s = d = wave - E; }
    float alpha = w[wave] / (denom[d] + 1e-16f);
    const float* hs = h + (size_t)s * F;
    float* od = out + (size_t)d * F;
    for (int f = lane; f < F; f += 32) atomicAdd(&od[f], alpha * hs[f]);
}

// Pass 3 (scalar features, layer 3).
__global__ void aggregate1(const int* __restrict__ ei, int E, int N,
                           const float* __restrict__ w, const float* __restrict__ denom,
                           const float* __restrict__ h3, float* __restrict__ out3) {
    int i = blockIdx.x * blockDim.x + threadIdx.x;
    int T = E + N;
    if (i >= T) return;
    int s, d;
    if (i < E) { s = ei[i]; d = ei[E + i]; } else { s = d = i - E; }
    float alpha = w[i] / (denom[d] + 1e-16f);
    atomicAdd(&out3[d], alpha * h3[s]);
}

// out += bias, then LeakyReLU(slope) in place.
__global__ void bias_act(float* __restrict__ out, const float* __restrict__ b, int N, int F,
                         float slope) {
    long i = blockIdx.x * (long)blockDim.x + threadIdx.x;
    if (i >= (long)N * F) return;
    int f = (int)(i % F);
    out[i] = leaky(out[i] + b[f], slope);
}

// Layer-3 GEMV + scalar attention logits: h3[n]=in[n,:]·W3; e_src=h3*a3s; e_dst=h3*a3d.
__global__ void gemv_att(const float* __restrict__ in, const float* __restrict__ W3,
                         const float* __restrict__ a3s, const float* __restrict__ a3d,
                         float* __restrict__ h3, float* __restrict__ esrc,
                         float* __restrict__ edst, int N, int K) {
    int wave = (int)((blockIdx.x * blockDim.x + threadIdx.x) >> 5);
    if (wave >= N) return;
    int lane = threadIdx.x & 31;
    const float* row = in + (size_t)wave * K;
    float s = 0.f;
    for (int f = lane; f < K; f += 32) s += row[f] * W3[f];
#pragma unroll
    for (int off = 16; off > 0; off >>= 1) s += __shfl_down(s, off, 32);
    if (lane == 0) {
        h3[wave] = s;
        esrc[wave] = s * a3s[0];
        edst[wave] = s * a3d[0];
    }
}

__global__ void final_out(const float* __restrict__ out3, const float* __restrict__ b3,
                          float* __restrict__ y, int N) {
    int i = blockIdx.x * blockDim.x + threadIdx.x;
    if (i < N) y[i] = out3[i] + b3[0];
}

static inline int gdiv(long n, int d) { return (int)((n + d - 1) / d); }

extern "C" void kernel_launch(void* const* d_in, const int* in_sizes, int n_in,
                              void* d_out, int out_size, void* d_ws, size_t ws_size,
                              hipStream_t stream) {
    const float* x   = (const float*)d_in[0];
    const int*   ei  = (const int*)d_in[1];
    const float* W1  = (const float*)d_in[2];
    const float* a1s = (const float*)d_in[3];
    const float* a1d = (const float*)d_in[4];
    const float* b1  = (const float*)d_in[5];
    const float* W2  = (const float*)d_in[6];
    const float* a2s = (const float*)d_in[7];
    const float* a2d = (const float*)d_in[8];
    const float* b2  = (const float*)d_in[9];
    const float* W3  = (const float*)d_in[10];
    const float* a3s = (const float*)d_in[11];
    const float* a3d = (const float*)d_in[12];
    const float* b3  = (const float*)d_in[13];

    const int N = in_sizes[0] / 8;      // 100000
    const int E = in_sizes[1] / 2;      // 1600000
    const int T = E + N;                // edges incl. self-loops

    float* ws    = (float*)d_ws;
    float* bufA  = ws;                              // N*256 (h1 / h2 / h3+out3)
    float* bufB  = bufA + (size_t)N * 256;          // N*256 (out1 / out2)
    float* esrc  = bufB + (size_t)N * 256;          // N
    float* edst  = esrc + N;                        // N
    float* m     = edst + N;                        // N
    float* denom = m + N;                           // N
    float* wE    = denom + N;                       // T

    const int BS = 256;                 // 8 waves per block

    // ---------------- Layer 1: Fin=8 -> Fout=256 ----------------
    {
        const int F = 256, K = 8;
        int ntiles = (N / 16) * (F / 16);
        gemm_wmma_f32<<<gdiv(ntiles, 8), BS, 0, stream>>>(x, W1, bufA, N, F, K);
        dot2_kernel<<<gdiv(N, 8), BS, 0, stream>>>(bufA, a1s, a1d, esrc, edst, N, F);
        fill_kernel<<<gdiv(N, BS), BS, 0, stream>>>(m, -FLT_MAX_F, N);
        fill_kernel<<<gdiv(N, BS), BS, 0, stream>>>(denom, 0.f, N);
        fill_kernel<<<gdiv((long)N * F, BS), BS, 0, stream>>>(bufB, 0.f, (long)N * F);
        edge_emax<<<gdiv(T, BS), BS, 0, stream>>>(ei, E, N, esrc, edst, wE, m);
        edge_expsum<<<gdiv(T, BS), BS, 0, stream>>>(ei, E, N, m, wE, denom);
        aggregate_vec<<<gdiv((long)T * 32, BS), BS, 0, stream>>>(ei, E, N, wE, denom, bufA,
                                                                 bufB, F);
        bias_act<<<gdiv((long)N * F, BS), BS, 0, stream>>>(bufB, b1, N, F, ACT_SLOPE);
    }

    // ---------------- Layer 2: 256 -> 128 ----------------
    {
        const int F = 128, K = 256;
        int ntiles = (N / 16) * (F / 16);
        gemm_wmma_f32<<<gdiv(ntiles, 8), BS, 0, stream>>>(bufB, W2, bufA, N, F, K);
        dot2_kernel<<<gdiv(N, 8), BS, 0, stream>>>(bufA, a2s, a2d, esrc, edst, N, F);
        fill_kernel<<<gdiv(N, BS), BS, 0, stream>>>(m, -FLT_MAX_F, N);
        fill_kernel<<<gdiv(N, BS), BS, 0, stream>>>(denom, 0.f, N);
        fill_kernel<<<gdiv((long)N * F, BS), BS, 0, stream>>>(bufB, 0.f, (long)N * F);
        edge_emax<<<gdiv(T, BS), BS, 0, stream>>>(ei, E, N, esrc, edst, wE, m);
        edge_expsum<<<gdiv(T, BS), BS, 0, stream>>>(ei, E, N, m, wE, denom);
        aggregate_vec<<<gdiv((long)T * 32, BS), BS, 0, stream>>>(ei, E, N, wE, denom, bufA,
                                                                 bufB, F);
        bias_act<<<gdiv((long)N * F, BS), BS, 0, stream>>>(bufB, b2, N, F, ACT_SLOPE);
    }

    // ---------------- Layer 3: 128 -> 1 ----------------
    {
        float* h3   = bufA;
        float* out3 = bufA + N;
        gemv_att<<<gdiv(N, 8), BS, 0, stream>>>(bufB, W3, a3s, a3d, h3, esrc, edst, N, 128);
        fill_kernel<<<gdiv(N, BS), BS, 0, stream>>>(m, -FLT_MAX_F, N);
        fill_kernel<<<gdiv(N, BS), BS, 0, stream>>>(denom, 0.f, N);
        fill_kernel<<<gdiv(N, BS), BS, 0, stream>>>(out3, 0.f, N);
        edge_emax<<<gdiv(T, BS), BS, 0, stream>>>(ei, E, N, esrc, edst, wE, m);
        edge_expsum<<<gdiv(T, BS), BS, 0, stream>>>(ei, E, N, m, wE, denom);
        aggregate1<<<gdiv(T, BS), BS, 0, stream>>>(ei, E, N, wE, denom, h3, out3);
        final_out<<<gdiv(N, BS), BS, 0, stream>>>(out3, b3, (float*)d_out, N);
    }
}